// CognitiveWorkspace_34299608825939
// MI455X (gfx1250) — compile-verified
//
#include <hip/hip_runtime.h>

typedef __attribute__((ext_vector_type(2))) float v2f;
typedef __attribute__((ext_vector_type(4))) float f4;
typedef __attribute__((ext_vector_type(8))) float v8f;

#define N_LAYERS      24
#define D_MODEL       2048
#define D_SPOKE       128
#define D_HUB_PRIV    64
#define D_HUB_SHARED  512
#define D_TAG         64
#define HUB_PRIV_OFF  3072
#define HUB_SHARED_OFF 4608
#define TAG_OFF       5120
#define D_S           6656
#define DECAY_WINDOW  8

__global__ __launch_bounds__(256)
void CognitiveWorkspace_kernel(const float* __restrict__ S,
                               const float* __restrict__ H,
                               const float* __restrict__ w_spoke,
                               const float* __restrict__ w_hub_priv,
                               const float* __restrict__ w_hub_shared,
                               const float* __restrict__ tagv,
                               const float* __restrict__ Wq,
                               const float* __restrict__ Wg,
                               const float* __restrict__ bg,
                               const int*   __restrict__ layer_idx,
                               float* __restrict__ outS,
                               float* __restrict__ outG)
{
    __shared__ float gate_in[16][128];  // [token][ query(0..63) | mean_tag(64..127) ]  8 KB
    __shared__ float gate[16][512];     // sigmoid gate tile                            32 KB

    const int li    = layer_idx[0];
    const int start = (li > DECAY_WINDOW) ? (li - DECAY_WINDOW) : 0;
    const int n_tags = li - start;
    const int ss  = li * D_SPOKE;                  // spoke slice start
    const int hps = HUB_PRIV_OFF + li * D_HUB_PRIV;
    const int ts  = TAG_OFF + li * D_TAG;

    const int bt0  = blockIdx.x * 16;              // first token of this tile
    const int tid  = threadIdx.x;
    const int wave = tid >> 5;
    const int lane = tid & 31;
    const int half = lane >> 4;                    // lane half (K split for A/B)
    const int l16  = lane & 15;

    // ---------------- Phase 1 ----------------
    if (wave < 4) {
        // query[16x64] = H[16x2048] x Wq^T ; this wave owns N-tile [wave*16, wave*16+16)
        // f32 WMMA A 16x4 layout: lanes 0-15 hold K=0,1 ; lanes 16-31 hold K=2,3
        const int m  = l16;
        const int n  = wave * 16 + l16;
        const int ka = half * 2;
        const float* hrow = H  + (size_t)(bt0 + m) * D_MODEL + ka;
        const float* qrow = Wq + (size_t)n * D_MODEL + ka;
        v8f acc = {};
#pragma unroll 8
        for (int k = 0; k < D_MODEL; k += 4) {
            v2f a = *(const v2f*)(hrow + k);
            v2f b = *(const v2f*)(qrow + k);
            acc = __builtin_amdgcn_wmma_f32_16x16x4_f32(
                false, a, false, b, (short)0, acc, false, false);
        }
        // D layout: VGPR r, halves -> row = r + 8*half, col = l16
#pragma unroll
        for (int r = 0; r < 8; ++r)
            gate_in[r + 8 * half][wave * 16 + l16] = acc[r];
    } else {
        // mean_tag: 16 tokens x 64 dims; 128 threads -> 8 outputs each
        // (regular-temporal loads: these bytes of S are re-read in phase 3)
        const int t2 = tid - 128;
        const float invn = (n_tags > 0) ? (1.0f / (float)n_tags) : 0.0f;
#pragma unroll
        for (int j = 0; j < 8; ++j) {
            int idx = j * 128 + t2;
            int row = idx >> 6;
            int dim = idx & 63;
            const float* p = S + (size_t)(bt0 + row) * D_S + TAG_OFF + start * D_TAG + dim;
            float acc = 0.0f;
            for (int tg = 0; tg < n_tags; ++tg) acc += p[tg * D_TAG];
            gate_in[row][64 + dim] = acc * invn;
        }
    }
    __syncthreads();

    // ---------------- Phase 2: gate = sigmoid(gate_in[16x128] x Wg^T + bg) ----------
    {
        const int m  = l16;
        const int ka = half * 2;
        v8f acc0 = {}, acc1 = {}, acc2 = {}, acc3 = {};
        const int nbase = wave * 64 + l16;         // 4 N-tiles per wave
        for (int k = 0; k < 2 * D_TAG; k += 4) {
            v2f a = *(const v2f*)(&gate_in[m][k + ka]);   // one ds_load_b64
            v2f b0 = *(const v2f*)(Wg + (size_t)(nbase +  0) * 128 + k + ka);
            v2f b1 = *(const v2f*)(Wg + (size_t)(nbase + 16) * 128 + k + ka);
            v2f b2 = *(const v2f*)(Wg + (size_t)(nbase + 32) * 128 + k + ka);
            v2f b3 = *(const v2f*)(Wg + (size_t)(nbase + 48) * 128 + k + ka);
            acc0 = __builtin_amdgcn_wmma_f32_16x16x4_f32(false, a, false, b0, (short)0, acc0, false, false);
            acc1 = __builtin_amdgcn_wmma_f32_16x16x4_f32(false, a, false, b1, (short)0, acc1, false, false);
            acc2 = __builtin_amdgcn_wmma_f32_16x16x4_f32(false, a, false, b2, (short)0, acc2, false, false);
            acc3 = __builtin_amdgcn_wmma_f32_16x16x4_f32(false, a, false, b3, (short)0, acc3, false, false);
        }
        v8f* accs[4] = {&acc0, &acc1, &acc2, &acc3};
#pragma unroll
        for (int t = 0; t < 4; ++t) {
            const int n = nbase + t * 16;
            const float bv = bg[n];
            v8f d = *accs[t];
#pragma unroll
            for (int r = 0; r < 8; ++r) {
                float v = d[r] + bv;
                v = 1.0f / (1.0f + __expf(-v));
                gate[r + 8 * half][n] = v;
            }
        }
    }
    __syncthreads();

    // ---------------- Phase 3: fused streaming update of S, write gate -------------
    // Stream-once traffic uses non-temporal policy so Wq/Wg stay L2-resident.
    // Regions start/end on multiples of 64 -> every aligned float4 is in one region.
    for (int i = tid; i < 16 * (D_S / 4); i += 256) {
        const int row = i / (D_S / 4);
        const int col = (i - row * (D_S / 4)) * 4;
        const size_t sidx = (size_t)(bt0 + row) * D_S + col;
        f4 v = __builtin_nontemporal_load((const f4*)(S + sidx));
        if (col >= HUB_SHARED_OFF && col < TAG_OFF) {
            const int g = col - HUB_SHARED_OFF;
            f4 gv = *(const f4*)(&gate[row][g]);
            f4 w  = __builtin_nontemporal_load(
                        (const f4*)(w_hub_shared + (size_t)(bt0 + row) * D_HUB_SHARED + g));
            v = v * gv + w;
        }
        if (col >= ss && col < ss + D_SPOKE) {
            f4 w = __builtin_nontemporal_load(
                       (const f4*)(w_spoke + (size_t)(bt0 + row) * D_SPOKE + (col - ss)));
            v = v + w;
        }
        if (col >= hps && col < hps + D_HUB_PRIV) {
            f4 w = __builtin_nontemporal_load(
                       (const f4*)(w_hub_priv + (size_t)(bt0 + row) * D_HUB_PRIV + (col - hps)));
            v = v + w;
        }
        if (col >= ts && col < ts + D_TAG) {
            f4 w = __builtin_nontemporal_load(
                       (const f4*)(tagv + (size_t)(bt0 + row) * D_TAG + (col - ts)));
            v = v + w;
        }
        __builtin_nontemporal_store(v, (f4*)(outS + sidx));
    }
    for (int i = tid; i < 16 * (D_HUB_SHARED / 4); i += 256) {
        const int row = i >> 7;
        const int c   = (i & 127) * 4;
        f4 gv = *(const f4*)(&gate[row][c]);
        __builtin_nontemporal_store(gv, (f4*)(outG + (size_t)(bt0 + row) * D_HUB_SHARED + c));
    }
}

extern "C" void kernel_launch(void* const* d_in, const int* in_sizes, int n_in,
                              void* d_out, int out_size, void* d_ws, size_t ws_size,
                              hipStream_t stream) {
    const float* S            = (const float*)d_in[0];
    const float* H            = (const float*)d_in[1];
    const float* w_spoke      = (const float*)d_in[2];
    const float* w_hub_priv   = (const float*)d_in[3];
    const float* w_hub_shared = (const float*)d_in[4];
    const float* tagv         = (const float*)d_in[5];
    const float* Wq           = (const float*)d_in[6];
    const float* Wg           = (const float*)d_in[7];
    const float* bg           = (const float*)d_in[8];
    const int*   layer_idx    = (const int*)d_in[9];

    const int BT = in_sizes[1] / D_MODEL;          // B*T from H's flat size
    float* outS = (float*)d_out;
    float* outG = outS + (size_t)BT * D_S;

    const int nblocks = BT / 16;
    CognitiveWorkspace_kernel<<<nblocks, 256, 0, stream>>>(
        S, H, w_spoke, w_hub_priv, w_hub_shared, tagv, Wq, Wg, bg, layer_idx,
        outS, outG);
}